// DualMarginContrastiveLoss_20091857011319
// MI455X (gfx1250) — compile-verified
//
#include <hip/hip_runtime.h>
#include <hip/hip_bf16.h>

// ---------------------------------------------------------------------------
// DualMarginContrastiveLoss on MI455X (gfx1250).
//   G = F * F^T via V_WMMA_F32_16X16X4_F32 (fp32 matrix pipe).
//   B panels staged in LDS with global_load_async_to_lds_b128 (ASYNCcnt),
//   double-buffered, shared by 8 waves per workgroup (8x less L2 traffic).
//   Margin/ReLU epilogue + deterministic two-stage reduction.
// ---------------------------------------------------------------------------

typedef __attribute__((ext_vector_type(2))) float v2f;
typedef __attribute__((ext_vector_type(4))) float v4f;
typedef __attribute__((ext_vector_type(8))) float v8f;

#define B_N      8192
#define D_K      256
#define T_TILES  (B_N / 16)          // 512 16-row tiles per side
#define IW       8                   // i-tiles per workgroup (1 per wave)
#define JW       8                   // j-tiles per chunk (shared via LDS)
#define GX       (T_TILES / IW)      // 64
#define GY       (T_TILES / JW)      // 64
#define NPART    (GX * GY * IW)      // 32768 wave partials
#define LDS_ROW  260                 // 256 floats + 4 pad -> bank-conflict-free
#define MARGIN1  5.0f
#define MARGIN2  10.0f

// --------------------------------------------------------------------------
// Kernel 1: squared L2 norm of each row of F. One wave per row.
// --------------------------------------------------------------------------
__global__ __launch_bounds__(256)
void rownorm_kernel(const float* __restrict__ F, float* __restrict__ sq) {
    const int wave = threadIdx.x >> 5;
    const int lane = threadIdx.x & 31;
    const int row  = blockIdx.x * 8 + wave;
    const float* p = F + (size_t)row * D_K + lane * 8;
    v4f a = *(const v4f*)(p + 0);
    v4f b = *(const v4f*)(p + 4);
    float s = a.x * a.x + a.y * a.y + a.z * a.z + a.w * a.w
            + b.x * b.x + b.y * b.y + b.z * b.z + b.w * b.w;
    #pragma unroll
    for (int off = 16; off > 0; off >>= 1)
        s += __shfl_down(s, off, 32);
    if (lane == 0) sq[row] = s;
}

// --------------------------------------------------------------------------
// Async cooperative fill of one 16x256 f32 B panel into LDS (padded rows).
// 256 threads x 4 chunks x 16B = 16 KB. Consecutive lanes -> consecutive
// 16B chunks (fully coalesced). Tracked with ASYNCcnt.
// --------------------------------------------------------------------------
__device__ __forceinline__
void fill_async(const float* __restrict__ F, int j0, float* dst, int tid) {
    #pragma unroll
    for (int it = 0; it < 4; ++it) {
        const int cidx = it * 256 + tid;           // 0..1023
        const int row  = cidx >> 6;                // 0..15
        const int c4   = cidx & 63;                // 16B chunk within row
        const uint64_t ga  =
            (uint64_t)(uintptr_t)(F + (size_t)(j0 + row) * D_K + c4 * 4);
        const uint32_t lds =
            (uint32_t)(uintptr_t)(dst + row * LDS_ROW + c4 * 4);
        asm volatile("global_load_async_to_lds_b128 %0, %1, off"
                     :: "v"(lds), "v"(ga) : "memory");
    }
}

// --------------------------------------------------------------------------
// Kernel 2: workgroup = 8 waves. Wave w computes i-tile (bx*8+w) against
// j-tiles by*8 .. by*8+7, whose B panels are staged in LDS and shared.
//   A frag (f32 16x4): lane r holds K=2*half,2*half+1 of row i0+r.
//   B frag mirrors it from the LDS panel (rows j0..j0+15, padded stride).
//   C/D: VGPR v <-> M=v+8*half, lane <-> N=lane%16.
// --------------------------------------------------------------------------
__global__ __launch_bounds__(256)
void tile_kernel(const float* __restrict__ F,
                 const int*   __restrict__ labels,
                 const int*   __restrict__ mlabels,
                 const float* __restrict__ sq,
                 float*       __restrict__ partials) {
    __shared__ float bbuf[2][16 * LDS_ROW];        // 2 x 16.6 KB, double buffer

    const int tid  = threadIdx.x;
    const int w    = tid >> 5;                     // wave id 0..7
    const int lane = tid & 31;
    const int half = lane >> 4;
    const int r    = lane & 15;

    const int ti0  = blockIdx.x * IW;              // first i-tile of block
    const int tj0  = blockIdx.y * JW;              // first j-tile of block
    const int ti   = ti0 + w;                      // this wave's i-tile
    const int i0   = ti * 16;
    const int pidx = (blockIdx.x * GY + blockIdx.y) * IW + w;

    if (tj0 + JW - 1 < ti0) {                      // block strictly below diag
        if (lane == 0) partials[pidx] = 0.0f;
        return;
    }

    // ---- A panel (16x256) into 128 VGPRs, held for all 8 j-tiles ----
    v2f afrag[D_K / 4];
    const float* rowA = F + (size_t)(i0 + r) * D_K + 2 * half;
    #pragma unroll
    for (int kk = 0; kk < D_K / 4; ++kk)
        afrag[kk] = *(const v2f*)(rowA + kk * 4);

    // ---- i-side metadata (fixed per wave) ----
    float sq_i[8]; int lab_i[8]; int ml_i[8];
    #pragma unroll
    for (int v = 0; v < 8; ++v) {
        const int i = i0 + v + 8 * half;
        sq_i[v]  = sq[i];
        lab_i[v] = labels[i];
        ml_i[v]  = mlabels[i];
    }

    float local = 0.0f;

    fill_async(F, tj0 * 16, &bbuf[0][0], tid);     // prologue fill

    #pragma unroll 1
    for (int t = 0; t < JW; ++t) {
        const int cur = t & 1;
        asm volatile("s_wait_asynccnt 0x0" ::: "memory");
        __syncthreads();                           // panel[cur] visible to all

        if (t + 1 < JW)                            // overlap next fill
            fill_async(F, (tj0 + t + 1) * 16, &bbuf[cur ^ 1][0], tid);

        const int tj = tj0 + t;
        if (tj >= ti) {
            const float* bp = &bbuf[cur][(size_t)r * LDS_ROW + 2 * half];

            v8f c = {};
            #pragma unroll
            for (int kk = 0; kk < D_K / 4; ++kk) {
                v2f b = *(const v2f*)(bp + kk * 4);    // ds_load_b64, no conflicts
                c = __builtin_amdgcn_wmma_f32_16x16x4_f32(
                        false, afrag[kk], false, b, (short)0, c, false, false);
            }

            // ---- epilogue: d2 = |Fi|^2 + |Fj|^2 - 2<Fi,Fj>, margins ----
            const int   j     = tj * 16 + r;
            const float sq_j  = sq[j];
            const int   lab_j = labels[j];
            const int   ml_j  = mlabels[j];
            const bool  off_diag = (tj > ti);

            #pragma unroll
            for (int v = 0; v < 8; ++v) {
                const int  m     = v + 8 * half;
                const bool valid = off_diag || (m < r);    // i < j only
                float d2 = sq_i[v] + sq_j - 2.0f * c[v];
                d2 = fmaxf(d2, 0.0f);
                const float dist = sqrtf(d2);
                float term;
                if (lab_i[v] == lab_j) {
                    term = dist;
                } else {
                    const float margin = (ml_i[v] == ml_j) ? MARGIN1 : MARGIN2;
                    term = fmaxf(margin - dist, 0.0f);
                }
                local += valid ? term : 0.0f;
            }
        }
        __syncthreads();                           // reads done before overwrite
    }

    // ---- wave reduce, one deterministic partial per wave ----
    #pragma unroll
    for (int off = 16; off > 0; off >>= 1)
        local += __shfl_down(local, off, 32);
    if (lane == 0) partials[pidx] = local;
}

// --------------------------------------------------------------------------
// Kernel 3: deterministic fixed-schedule reduction of all wave partials.
// --------------------------------------------------------------------------
__global__ __launch_bounds__(256)
void reduce_kernel(const float* __restrict__ partials, float* __restrict__ out) {
    __shared__ float smem[256];
    float s = 0.0f;
    for (int i = threadIdx.x; i < NPART; i += 256)
        s += partials[i];
    smem[threadIdx.x] = s;
    __syncthreads();
    #pragma unroll
    for (int stride = 128; stride > 0; stride >>= 1) {
        if (threadIdx.x < stride)
            smem[threadIdx.x] += smem[threadIdx.x + stride];
        __syncthreads();
    }
    if (threadIdx.x == 0) {
        const double n_pairs = (double)B_N * (double)(B_N - 1) * 0.5;
        out[0] = (float)((double)smem[0] / n_pairs);
    }
}

// --------------------------------------------------------------------------
extern "C" void kernel_launch(void* const* d_in, const int* in_sizes, int n_in,
                              void* d_out, int out_size, void* d_ws, size_t ws_size,
                              hipStream_t stream) {
    const float* F       = (const float*)d_in[0];   // [8192, 256] fp32
    const int*   labels  = (const int*)d_in[1];     // [8192]
    const int*   mlabels = (const int*)d_in[2];     // [8192]
    float*       out     = (float*)d_out;           // scalar fp32

    float* ws       = (float*)d_ws;
    float* sq       = ws;                            // 8192 floats
    float* partials = ws + B_N;                      // 32768 floats

    rownorm_kernel<<<B_N / 8, 256, 0, stream>>>(F, sq);

    dim3 grid(GX, GY);
    tile_kernel<<<grid, 256, 0, stream>>>(F, labels, mlabels, sq, partials);

    reduce_kernel<<<1, 256, 0, stream>>>(partials, out);
}